// StaticGraphAttentionLayer_26147760898458
// MI455X (gfx1250) — compile-verified
//
#include <hip/hip_runtime.h>

// fp32 WMMA fragment types (wave32): A/B = 2 VGPRs, C/D = 8 VGPRs
typedef __attribute__((ext_vector_type(2))) float v2f;
typedef __attribute__((ext_vector_type(8))) float v8f;
// TDM descriptor word types
typedef unsigned int u32x4 __attribute__((ext_vector_type(4)));
typedef int i32x4 __attribute__((ext_vector_type(4)));
typedef int i32x8 __attribute__((ext_vector_type(8)));

// Problem constants (fixed by the reference)
#define B_  16
#define S_  512
#define T_  64
#define L_  64
#define U_  128

// LDS layout (floats). te/ht padded to stride 132 (3*132 mod 64 = 12 -> spreads banks).
#define TS 132
#define HS 132
#define TE_OFF 0                         // 192 x TS
#define HT_OFF (192 * TS)                // 25344
#define EW_OFF (HT_OFF + 64 * HS)        // 33792
#define AL_OFF (EW_OFF + 64)             // 33856
#define WT_OFF (AL_OFF + 64)             // 33920 : W_trans packed pairs (64x128 f2)
#define WR_OFF (WT_OFF + 128 * 128)      // 50304 : W_rel packed pairs   (64x128 f2)
#define SMEM_FLOATS (WR_OFF + 128 * 128) // 66688 floats = 266752 B (< 320KB/WG)

// d_ws layout (floats): ge, then packed weight pair arrays
#define GE_FLOATS  (B_ * T_ * 256)       // 262144
#define WHP_OFF    GE_FLOATS             // W_ht pairs: 128x128 float2 = 32768 f
#define WTP_OFF    (WHP_OFF + 32768)     // W_trans pairs: 64x128 float2 = 16384 f
#define WRP_OFF    (WTP_OFF + 16384)     // W_rel pairs:   64x128 float2 = 16384 f

__device__ __forceinline__ float fast_tanh(float x) {
#if __has_builtin(__builtin_amdgcn_tanhf)
  return __builtin_amdgcn_tanhf(x);      // V_TANH_F32 (gfx1250 trans op)
#else
  float e = __expf(2.0f * x);            // branch-free fallback
  return 1.0f - 2.0f / (e + 1.0f);
#endif
}

__device__ __forceinline__ v8f wmma4(v2f a, v2f b, v8f c) {
  // D = A(16x4,f32) x B(4x16,f32) + C(16x16,f32)
  return __builtin_amdgcn_wmma_f32_16x16x4_f32(false, a, false, b, (short)0, c,
                                               false, false);
}

#if __has_builtin(__builtin_amdgcn_tensor_load_to_lds)
#define HAVE_TDM 1
// Issue a 2D TDM load: global (dim1 rows x dim0 f32, row stride stride0) -> LDS.
__device__ __forceinline__ void tdm_load_2d(unsigned int lds_byte_off,
                                            const void* gptr,
                                            unsigned int dim0, unsigned int dim1,
                                            unsigned int stride0) {
  unsigned long long ga = (unsigned long long)(uintptr_t)gptr;
  u32x4 g0;
  g0.x = 1u;                                            // count=1, user-mode D#
  g0.y = lds_byte_off;                                  // lds_addr
  g0.z = (unsigned int)ga;                              // global_addr[31:0]
  g0.w = (unsigned int)((ga >> 32) & 0x01FFFFFFu)       // global_addr[56:32]
         | (2u << 30);                                  // type = 2 ("image")
  i32x8 g1;
  g1[0] = (int)(2u << 16);                              // data_size = 4 bytes
  g1[1] = (int)(dim0 << 16);                            // tensor_dim0[15:0]
  g1[2] = (int)((dim0 >> 16) | (dim1 << 16));           // dim0 hi | tensor_dim1 lo
  g1[3] = (int)((dim1 >> 16) | (dim0 << 16));           // dim1 hi | tile_dim0
  g1[4] = (int)(dim1 & 0xFFFFu);                        // tile_dim1 (tile_dim2=0)
  g1[5] = (int)stride0;                                 // tensor_dim0_stride lo
  g1[6] = 0;
  g1[7] = 0;
  i32x4 z4 = {0, 0, 0, 0};
#if __has_include(<hip/amd_detail/amd_gfx1250_TDM.h>)
  i32x8 z8 = {};                                        // clang-23 6-arg form
  __builtin_amdgcn_tensor_load_to_lds(g0, g1, z4, z4, z8, 0);
#else
  __builtin_amdgcn_tensor_load_to_lds(g0, g1, z4, z4, 0);  // ROCm 7.2 5-arg form
#endif
}
#else
#define HAVE_TDM 0
#endif

__global__ __launch_bounds__(384)
void graph_attn_fused(const int* __restrict__ triples,      // (B,T,L,3)
                      const float* __restrict__ emb,        // (VOCAB,128)
                      const float2* __restrict__ Wtp,       // W_trans pairs 64x128
                      const float* __restrict__ bt_,        // (128)
                      const float2* __restrict__ Whp,       // W_ht pairs 128x128
                      const float* __restrict__ bht,        // (128)
                      const float2* __restrict__ Wrp,       // W_rel pairs 64x128
                      const float* __restrict__ br,         // (128)
                      float* __restrict__ ge) {             // (B*T,256)
  extern __shared__ float smem[];
  float* te  = smem + TE_OFF;   // 192 x 128 (stride TS)
  float* ht  = smem + HT_OFF;   // 64 x 128 (stride HS)
  float* ew  = smem + EW_OFF;   // 64
  float* al  = smem + AL_OFF;   // 64
  const v2f* WtLp = (const v2f*)(smem + WT_OFF);  // packed pairs in LDS
  const v2f* WrLp = (const v2f*)(smem + WR_OFF);

  const int bt   = blockIdx.x;          // b*T + t
  const int tid  = threadIdx.x;
  const int wave = tid >> 5;            // 0..11 (wave32)
  const int lane = tid & 31;
  const int l16  = lane & 15;
  const int half = lane >> 4;           // A: K-half select, C: M +8 offset

  if (tid < 64) ew[tid] = 0.0f;

  // ---------------- Phase A: TDM-stage packed W_trans & W_rel into LDS -------
#if HAVE_TDM
  if (wave == 0) {                      // one TDM issue per descriptor per WG
    tdm_load_2d(WT_OFF * 4u, Wtp, 16384u, 1u, 16384u);  // 64KB contiguous
    tdm_load_2d(WR_OFF * 4u, Wrp, 16384u, 1u, 16384u);
    __builtin_amdgcn_s_wait_tensorcnt(0);
  }
#else
  for (int i = tid; i < 4096; i += 384)
    ((float4*)(smem + WT_OFF))[i] = ((const float4*)Wtp)[i];
  for (int i = tid; i < 4096; i += 384)
    ((float4*)(smem + WR_OFF))[i] = ((const float4*)Wrp)[i];
#endif
  __syncthreads();

  // ---------------- Phase B: te = tanh(E @ Wt + bt), 192 rows ----------------
  // A-fragments gathered straight from the L2-resident embedding table;
  // each element consumed exactly once (8 C-tiles live in registers per wave).
  {
    const int row  = wave * 16 + l16;
    const int vidx = triples[bt * 192 + row];
    const float* erow = emb + (size_t)vidx * U_;
    v8f z = {};
    v8f acc[8];
#pragma unroll
    for (int n = 0; n < 8; ++n) acc[n] = z;
    for (int k0 = 0; k0 < 128; k0 += 4) {
      const int kk = k0 + 2 * half;
      const int kp = (k0 >> 1) + half;          // (kk,kk+1) pair index
      v2f a = *(const v2f*)(erow + kk);
#pragma unroll
      for (int n = 0; n < 8; ++n) {
        v2f b = WtLp[kp * U_ + n * 16 + l16];   // single ds_load_b64
        acc[n] = wmma4(a, b, acc[n]);
      }
    }
#pragma unroll
    for (int n = 0; n < 8; ++n) {
      const int col = n * 16 + l16;
      const float bias = bt_[col];
#pragma unroll
      for (int r = 0; r < 8; ++r) {
        const int m = wave * 16 + r + 8 * half;
        te[m * TS + col] = fast_tanh(acc[n][r] + bias);
      }
    }
  }
  __syncthreads();

  // ---------------- Phase C: ht GEMM (waves 0-7) / rel GEMM (waves 8-11) -----
  v8f relacc[8];
  {
    v8f z = {};
#pragma unroll
    for (int n = 0; n < 8; ++n) relacc[n] = z;
  }
  int relmb = 0;
  if (wave < 8) {
    // ht_t = tanh([head||tail] @ Wht + bht). wave -> (mb = w>>1, 64-col half)
    const int mb    = wave >> 1;
    const int nbase = (wave & 1) * 64;
    const int lrow  = mb * 16 + l16;
    const float* headrow = te + (3 * lrow) * TS;
    const float* tailrow = te + (3 * lrow + 2) * TS;
    v8f z = {};
    v8f acc[4];
#pragma unroll
    for (int n = 0; n < 4; ++n) acc[n] = z;
    for (int k0 = 0; k0 < 256; k0 += 4) {
      const int kk = k0 + 2 * half;  // kk,kk+1 stay on the same side of 128
      v2f a = (kk < 128) ? *(const v2f*)(headrow + kk)
                         : *(const v2f*)(tailrow + (kk - 128));
      const int kp = (k0 >> 1) + half;   // packed (kk,kk+1) pair index
#pragma unroll
      for (int n = 0; n < 4; ++n) {
        const int col = nbase + n * 16 + l16;
        v2f b = *(const v2f*)(Whp + kp * U_ + col);  // one global_load_b64
        acc[n] = wmma4(a, b, acc[n]);
      }
    }
#pragma unroll
    for (int n = 0; n < 4; ++n) {
      const int col = nbase + n * 16 + l16;
      const float bias = bht[col];
#pragma unroll
      for (int r = 0; r < 8; ++r) {
        const int m = mb * 16 + r + 8 * half;
        ht[m * HS + col] = fast_tanh(acc[n][r] + bias);
      }
    }
  } else {
    // rel_t = REL @ Wr + br, kept in registers. wave 8..11 -> mb 0..3
    const int mb = wave - 8;
    relmb = mb;
    const int lrow = mb * 16 + l16;
    const float* relrow = te + (3 * lrow + 1) * TS;
    for (int k0 = 0; k0 < 128; k0 += 4) {
      const int kk = k0 + 2 * half;
      const int kp = (k0 >> 1) + half;
      v2f a = *(const v2f*)(relrow + kk);
#pragma unroll
      for (int n = 0; n < 8; ++n) {
        v2f b = WrLp[kp * U_ + n * 16 + l16];   // single ds_load_b64
        relacc[n] = wmma4(a, b, relacc[n]);
      }
    }
#pragma unroll
    for (int n = 0; n < 8; ++n) {
      const float bias = br[n * 16 + l16];
#pragma unroll
      for (int r = 0; r < 8; ++r) relacc[n][r] += bias;
    }
  }
  __syncthreads();

  // ---------------- Phase D: e_weight[l] = dot(rel_t[l], ht_t[l]) ------------
  if (wave >= 8) {
#pragma unroll
    for (int r = 0; r < 8; ++r) {
      const int row = relmb * 16 + r + 8 * half;
      float p = 0.0f;
#pragma unroll
      for (int n = 0; n < 8; ++n) {
        const int col = n * 16 + l16;
        p += relacc[n][r] * ht[row * HS + col];
      }
      atomicAdd(&ew[row], p);   // ds_add_f32
    }
  }
  __syncthreads();

  // ---------------- Phase E: softmax over L=64 (wave 0, wave32 shuffles) -----
  if (wave == 0) {
    float v0 = ew[lane], v1 = ew[lane + 32];
    float m = fmaxf(v0, v1);
#pragma unroll
    for (int off = 16; off > 0; off >>= 1)
      m = fmaxf(m, __shfl_xor(m, off, 32));
    float e0 = __expf(v0 - m), e1 = __expf(v1 - m);
    float s = e0 + e1;
#pragma unroll
    for (int off = 16; off > 0; off >>= 1)
      s += __shfl_xor(s, off, 32);
    al[lane]      = e0 / s;
    al[lane + 32] = e1 / s;
  }
  __syncthreads();

  // ---------------- Phase F: graph_embed = sum_l alpha[l]*head_tail[l] -------
  if (tid < 256) {
    const int c = tid;
    float acc = 0.0f;
    if (c < 128) {
      for (int l = 0; l < 64; ++l) acc += al[l] * te[(3 * l) * TS + c];
    } else {
      const int cc = c - 128;
      for (int l = 0; l < 64; ++l) acc += al[l] * te[(3 * l + 2) * TS + cc];
    }
    ge[(size_t)bt * 256 + c] = acc;
  }
}

// Repack a row-major (2*npairs x 128) weight matrix into (k-pair, col) float2.
__global__ __launch_bounds__(256)
void pack_pairs(const float* __restrict__ W, float2* __restrict__ Wp, int npairs) {
  const int tid = blockIdx.x * blockDim.x + threadIdx.x;
  if (tid >= npairs * U_) return;
  const int k2  = tid >> 7;
  const int col = tid & 127;
  Wp[tid] = make_float2(W[(2 * k2) * U_ + col], W[(2 * k2 + 1) * U_ + col]);
}

// out[b,s,:] = graph_embed[b, sent_triples[b,s,0], :]
__global__ __launch_bounds__(256)
void gather_out(const int* __restrict__ sent_triples,   // (B,S,1)
                const float* __restrict__ ge,           // (B*T,256)
                float* __restrict__ out) {              // (B,S,256)
  const int gid = blockIdx.x * blockDim.x + threadIdx.x; // B*S*256 total
  const int c   = gid & 255;
  const int bs  = gid >> 8;        // b*S + s
  const int b   = bs >> 9;         // S = 512
  const int idx = sent_triples[bs];
  out[gid] = ge[((size_t)(b * T_ + idx)) * 256 + c];
}

extern "C" void kernel_launch(void* const* d_in, const int* in_sizes, int n_in,
                              void* d_out, int out_size, void* d_ws, size_t ws_size,
                              hipStream_t stream) {
  // d_in order: sents(0,unused), triples(1), sent_triples(2), emb_table(3),
  //             W_trans(4), b_trans(5), W_ht(6), b_ht(7), W_rel(8), b_rel(9)
  const int*   triples      = (const int*)  d_in[1];
  const int*   sent_triples = (const int*)  d_in[2];
  const float* emb          = (const float*)d_in[3];
  const float* Wt           = (const float*)d_in[4];
  const float* bt           = (const float*)d_in[5];
  const float* Wht          = (const float*)d_in[6];
  const float* bht          = (const float*)d_in[7];
  const float* Wr           = (const float*)d_in[8];
  const float* br           = (const float*)d_in[9];

  float*  ws  = (float*)d_ws;
  float*  ge  = ws;                         // B*T*256 floats
  float2* Whp = (float2*)(ws + WHP_OFF);    // 16384 float2
  float2* Wtp = (float2*)(ws + WTP_OFF);    // 8192 float2
  float2* Wrp = (float2*)(ws + WRP_OFF);    // 8192 float2
  float*  out = (float*)d_out;              // B*S*256 floats

  hipLaunchKernelGGL(pack_pairs, dim3(64), dim3(256), 0, stream, Wht, Whp, 128);
  hipLaunchKernelGGL(pack_pairs, dim3(32), dim3(256), 0, stream, Wt,  Wtp, 64);
  hipLaunchKernelGGL(pack_pairs, dim3(32), dim3(256), 0, stream, Wr,  Wrp, 64);

  const size_t smem = (size_t)SMEM_FLOATS * sizeof(float);
  hipLaunchKernelGGL(graph_attn_fused, dim3(B_ * T_), dim3(384), smem, stream,
                     triples, emb, Wtp, bt, Whp, bht, Wrp, br, ge);

  const int total = B_ * S_ * 256;
  hipLaunchKernelGGL(gather_out, dim3(total / 256), dim3(256), 0, stream,
                     sent_triples, ge, out);
}